// RegL1Loss_13821204758604
// MI455X (gfx1250) — compile-verified
//
#include <hip/hip_runtime.h>

// Problem constants from the reference:
//   output:       [64, 2, 512, 512] f32   (NCHW)
//   kp_projs_dis: [64, 128, 2]      f32
//   cord:         [64, 128, 2]      i32   (x = cord[...,0], y = cord[...,1])
//   result: mean SmoothL1 over 64*128*2 = 16384 elements (scalar f32)

#define BB        64
#define NKP       128
#define HW        512
#define PLANE     (512 * 512)      // H*W
#define NPAIRS    (BB * NKP)       // 8192 (b,kp) pairs
#define NELEM     (NPAIRS * 2)     // 16384 loss terms
#define BLK       256
#define NBLOCKS   (NPAIRS / BLK)   // 32

typedef __attribute__((ext_vector_type(2))) float v2f;
typedef __attribute__((ext_vector_type(8))) float v8f;

// Wave32 sum-reduction via V_WMMA_F32_16X16X4_F32.
// Lane L's partial is placed in A[L%16][0] (lanes 0-15) / A[L%16][2] (lanes 16-31),
// the other A column VGPR is zero.  B = ones(4x16), C = 0:
//   D[m][n] = rowsum_m = partial[m] + partial[m+16]   (replicated over n)
// Each lane's 8-VGPR D fragment therefore sums rows 0-7 (lanes 0-15: partials of lanes 0-7,16-23)
// or rows 8-15 (lanes 16-31: partials of lanes 8-15,24-31); one xor-16 shuffle-add gives the full
// wave total in every lane.  Requires EXEC all-ones (guaranteed by launch).
__device__ __forceinline__ float wave_reduce_wmma(float partial) {
    v2f a;
    a.x = partial;
    a.y = 0.0f;
    v2f bones;
    bones.x = 1.0f;
    bones.y = 1.0f;
    v8f c = {};
    // 8 args: (neg_a, A, neg_b, B, c_mod, C, reuse_a, reuse_b)
    c = __builtin_amdgcn_wmma_f32_16x16x4_f32(
        /*neg_a=*/false, a, /*neg_b=*/false, bones,
        /*c_mod=*/(short)0, c, /*reuse_a=*/false, /*reuse_b=*/false);
    float s = c[0] + c[1] + c[2] + c[3] + c[4] + c[5] + c[6] + c[7];
    s += __shfl_xor(s, 16, 32);
    return s;
}

__device__ __forceinline__ float smooth_l1(float pred, float tgt) {
    float d = fabsf(pred - tgt);
    return (d < 1.0f) ? (0.5f * d * d) : (d - 0.5f);
}

// Stage 1: one thread per (b, kp) pair.  Gathers both channels, computes the
// two SmoothL1 terms, WMMA-reduces per wave, LDS-combines 8 waves per block,
// writes one partial per block to the workspace.
__global__ __launch_bounds__(BLK) void RegL1Loss_stage1(
    const float* __restrict__ out_nchw,   // [64,2,512,512]
    const float* __restrict__ kp_dis,     // [64,128,2]
    const int*   __restrict__ cord,       // [64,128,2]
    float*       __restrict__ block_part) // [NBLOCKS]
{
    const int gid = blockIdx.x * BLK + threadIdx.x;   // 0 .. NPAIRS-1, exact
    const int b   = gid >> 7;                         // gid / NKP

    // Vectorized 8-byte loads (global_load_b64): coords + targets.
    const int2   cxy = ((const int2*)cord)[gid];      // {x, y}
    const float2 tgt = ((const float2*)kp_dis)[gid];  // {tx, ty}

    // output[b, c, y, x]; c=0 and c=1 planes are PLANE apart.
    const int base = (b * 2) * PLANE + cxy.y * HW + cxy.x;
    const float p0 = out_nchw[base];
    const float p1 = out_nchw[base + PLANE];

    const float partial = smooth_l1(p0, tgt.x) + smooth_l1(p1, tgt.y);

    // Wave-level matrix-op reduction (no divergence up to this point).
    const float wtot = wave_reduce_wmma(partial);

    __shared__ float lds[BLK / 32];
    const int lane = threadIdx.x & 31;
    const int wave = threadIdx.x >> 5;
    if (lane == 0) lds[wave] = wtot;
    __syncthreads();

    if (threadIdx.x == 0) {
        float s = 0.0f;
#pragma unroll
        for (int i = 0; i < BLK / 32; ++i) s += lds[i];
        block_part[blockIdx.x] = s;   // deterministic fixed-order sum
    }
}

// Stage 2: one wave32.  Lane i holds block partial i (NBLOCKS == 32),
// WMMA-reduce to the grand total, scale by 1/16384 (exact), write scalar.
__global__ __launch_bounds__(32) void RegL1Loss_stage2(
    const float* __restrict__ block_part,
    float*       __restrict__ result)
{
    const float v   = block_part[threadIdx.x];
    const float tot = wave_reduce_wmma(v);
    if (threadIdx.x == 0) {
        result[0] = tot * (1.0f / (float)NELEM);
    }
}

extern "C" void kernel_launch(void* const* d_in, const int* in_sizes, int n_in,
                              void* d_out, int out_size, void* d_ws, size_t ws_size,
                              hipStream_t stream) {
    const float* out_nchw = (const float*)d_in[0];  // [64,2,512,512] f32
    const float* kp_dis   = (const float*)d_in[1];  // [64,128,2]     f32
    const int*   cord     = (const int*)  d_in[2];  // [64,128,2]     i32
    float*       result   = (float*)d_out;          // scalar
    float*       scratch  = (float*)d_ws;           // >= NBLOCKS floats

    RegL1Loss_stage1<<<NBLOCKS, BLK, 0, stream>>>(out_nchw, kp_dis, cord, scratch);
    RegL1Loss_stage2<<<1, 32, 0, stream>>>(scratch, result);
}